// CausalBoW_48679159332942
// MI455X (gfx1250) — compile-verified
//
#include <hip/hip_runtime.h>
#include <stdint.h>

// CausalBoW: y[b,t,:] = cumsum_t(x) / (t+1).  B=8, T=4096, C=1024, fp32.
// Memory-bound (268 MB min traffic -> ~11.5us @ 23.3 TB/s). 3-phase chunked
// scan; phase 3 streams x through LDS via gfx1250 async global->LDS loads.
// x (134 MB) fits in the 192 MB L2: phase 1 warms it, phase 3 re-reads hit,
// y is written with non-temporal hints so it doesn't evict x.

typedef float vf4 __attribute__((ext_vector_type(4)));

#define B_      8
#define T_      4096
#define C_      1024
#define NCHUNK  64
#define TCHUNK  (T_ / NCHUNK)      // 64 t-steps per chunk
#define STAGE_T 4                  // t-rows per LDS stage (4 * 4KB = 16KB)
#define NSTAGES (TCHUNK / STAGE_T) // 16

#if defined(__HIP_DEVICE_COMPILE__) && __has_builtin(__builtin_amdgcn_global_load_async_to_lds_b128)
#define HAVE_ASYNC_BUILTIN 1
#else
#define HAVE_ASYNC_BUILTIN 0
#endif

#if defined(__HIP_DEVICE_COMPILE__) && __has_builtin(__builtin_amdgcn_s_wait_asynccnt)
#define HAVE_WAIT_BUILTIN 1
#else
#define HAVE_WAIT_BUILTIN 0
#endif

// One 16-byte async global->LDS transfer for this lane.
__device__ __forceinline__ void async_load_b128(const float* gp, float* lp,
                                                const float* gbase) {
#if HAVE_ASYNC_BUILTIN
  typedef int v4i_t __attribute__((vector_size(16)));
  typedef __attribute__((address_space(1))) v4i_t* gptr_t;  // global
  typedef __attribute__((address_space(3))) v4i_t* lptr_t;  // LDS (32-bit)
  // Integer round-trip avoids cross-AS pointer-cast restrictions; truncation
  // to 32 bits for AS3 yields the LDS byte offset (aperture low bits).
  __builtin_amdgcn_global_load_async_to_lds_b128(
      (gptr_t)(uintptr_t)gp,
      (lptr_t)(unsigned)(uintptr_t)lp,
      /*imm offset*/ 0, /*cpol*/ 0);
  (void)gbase;
#else
  unsigned lds_addr = (unsigned)(size_t)lp;
  unsigned goff     = (unsigned)((const char*)gp - (const char*)gbase);
  asm volatile("global_load_async_to_lds_b128 %0, %1, %2"
               :: "v"(lds_addr), "v"(goff), "s"(gbase)
               : "memory");
#endif
}

template <int N>
__device__ __forceinline__ void wait_asynccnt() {
#if HAVE_WAIT_BUILTIN
  __builtin_amdgcn_s_wait_asynccnt(N);
  asm volatile("" ::: "memory");   // keep LDS reads after the wait
#else
  asm volatile("s_wait_asynccnt %0" :: "n"(N) : "memory");
#endif
}

// ---------------- Phase 3: local cumsum + offset, via async LDS pipeline -----
// (placed first in the file so the disasm snippet shows the async codegen)
__global__ __launch_bounds__(256) void bow_apply(const float* __restrict__ x,
                                                 const float* __restrict__ offs,
                                                 float* __restrict__ y) {
  __shared__ float lds[2][STAGE_T][C_];   // 2 * 16KB double buffer
  const int c4    = threadIdx.x;
  const int chunk = blockIdx.x;
  const int b     = blockIdx.y;
  const int tg0   = chunk * TCHUNK;
  const size_t base = ((size_t)b * T_ + (size_t)tg0) * C_ + 4 * (size_t)c4;
  const float* xb = x + base;
  float*       yb = y + base;

  vf4 acc = ((const vf4*)offs)[((size_t)b * NCHUNK + chunk) * (C_ / 4) + c4];

  auto issue = [&](int s, int buf) {
#pragma unroll
    for (int i = 0; i < STAGE_T; ++i)
      async_load_b128(xb + (size_t)(s * STAGE_T + i) * C_, &lds[buf][i][4 * c4], x);
  };

  issue(0, 0);
  for (int s = 0; s < NSTAGES; ++s) {
    const int buf = s & 1;
    if (s + 1 < NSTAGES) {
      issue(s + 1, buf ^ 1);       // keep 2*STAGE_T B128s in flight per lane
      wait_asynccnt<STAGE_T>();    // stage s complete (next stage still pending)
    } else {
      wait_asynccnt<0>();
    }
    // Each lane consumes exactly the LDS bytes it requested -> wave-local,
    // no workgroup barrier needed.
#pragma unroll
    for (int i = 0; i < STAGE_T; ++i) {
      vf4 v = *(const vf4*)&lds[buf][i][4 * c4];
      acc += v;
      const int t = tg0 + s * STAGE_T + i;
      vf4 outv = acc / (float)(t + 1);
      __builtin_nontemporal_store(outv, (vf4*)(yb + (size_t)(s * STAGE_T + i) * C_));
    }
  }
}

// ---------------- Phase 1: per-chunk channel sums -> partial[b][chunk][c] ----
__global__ __launch_bounds__(256) void bow_chunk_sums(const float* __restrict__ x,
                                                      float* __restrict__ partial) {
  const int c4    = threadIdx.x;   // owns channels 4*c4 .. 4*c4+3
  const int chunk = blockIdx.x;
  const int b     = blockIdx.y;
  const vf4* p = (const vf4*)x + ((size_t)b * T_ + (size_t)chunk * TCHUNK) * (C_ / 4) + c4;
  vf4 acc = {0.f, 0.f, 0.f, 0.f};
#pragma unroll 8
  for (int t = 0; t < TCHUNK; ++t)
    acc += p[(size_t)t * (C_ / 4)];     // coalesced; RT hint warms L2 with x
  ((vf4*)partial)[((size_t)b * NCHUNK + chunk) * (C_ / 4) + c4] = acc;
}

// ---------------- Phase 2: exclusive scan over chunk sums (in registers) -----
// One thread per scalar column (b,c): preload all 64 chunk sums with
// independent (clause-able) loads -> one memory round-trip, scan in registers,
// store back in place. Avoids a serialized load->wait->store chain.
__global__ __launch_bounds__(256) void bow_exclusive_scan(float* __restrict__ partial) {
  const int g = blockIdx.x * 256 + threadIdx.x;   // 0 .. B_*C_-1
  const int b = g >> 10;                          // /C_
  const int c = g & (C_ - 1);
  float* p = partial + (size_t)b * NCHUNK * C_ + c;
  float v[NCHUNK];
#pragma unroll
  for (int k = 0; k < NCHUNK; ++k) v[k] = p[(size_t)k * C_];
  float run = 0.f;
#pragma unroll
  for (int k = 0; k < NCHUNK; ++k) {
    float t = v[k];
    p[(size_t)k * C_] = run;
    run += t;
  }
}

extern "C" void kernel_launch(void* const* d_in, const int* in_sizes, int n_in,
                              void* d_out, int out_size, void* d_ws, size_t ws_size,
                              hipStream_t stream) {
  const float* x = (const float*)d_in[0];
  float*       y = (float*)d_out;
  float* partial = (float*)d_ws;          // B_*NCHUNK*C_ floats = 2 MB scratch

  dim3 grid(NCHUNK, B_);                  // 512 blocks x 256 threads (8 waves each)
  bow_chunk_sums<<<grid, 256, 0, stream>>>(x, partial);
  bow_exclusive_scan<<<dim3((B_ * C_) / 256), 256, 0, stream>>>(partial);
  bow_apply<<<grid, 256, 0, stream>>>(x, partial, y);

  (void)in_sizes; (void)n_in; (void)out_size; (void)ws_size;
}